// PCModule_20194936226448
// MI455X (gfx1250) — compile-verified
//
#include <hip/hip_runtime.h>
#include <math.h>

#define CCH    128
#define NPIX   (512 * 512)
#define NBATCH 8
#define EPSF   1e-12f

typedef __attribute__((ext_vector_type(2))) float v2f;
typedef __attribute__((ext_vector_type(4))) float v4f;
typedef __attribute__((ext_vector_type(8))) float v8f;

__device__ __forceinline__ float wave_sum(float v) {
#pragma unroll
    for (int o = 16; o > 0; o >>= 1) v += __shfl_down(v, o, 32);
    return v;
}

// ---------------- kernel 0: zero the accumulator region of the workspace ----
__global__ void k_zero(float* __restrict__ ws, int n) {
    int i = blockIdx.x * blockDim.x + threadIdx.x;
    if (i < n) ws[i] = 0.f;
}

// ---------------- kernel 1: masked per-channel sums (prototype numerators) --
// grid = (NPIX/CHUNK, B), block = 256 (8 waves). gt mask staged in LDS once per
// pixel chunk; feats streamed once with coalesced 128-bit NT loads.
__global__ void k_reduce(const float* __restrict__ feats, const int* __restrict__ gt,
                         float* __restrict__ pcs, float* __restrict__ pns,
                         float* __restrict__ ncs) {
    const int CHUNK = 2048;
    int b    = blockIdx.y;
    int base = blockIdx.x * CHUNK;
    int tid  = threadIdx.x;
    int lane = tid & 31, wave = tid >> 5;

    __shared__ float mk[2048];
    __shared__ float sC[8], sT[8];

    // gt mask -> LDS, count ones (gt is {0,1} so n_n = NPIX - n_c)
    const int* g = gt + (size_t)b * NPIX + base;
    float cnt = 0.f;
    for (int i = tid; i < CHUNK; i += 256) {
        float m = (g[i] == 1) ? 1.f : 0.f;
        mk[i] = m;
        cnt += m;
    }
    float wcnt = wave_sum(cnt);
    if (lane == 0) sC[wave] = wcnt;
    __syncthreads();
    if (tid == 0) {
        float s = 0.f;
        for (int i = 0; i < 8; ++i) s += sC[i];
        atomicAdd(&ncs[b], s);
    }
    __syncthreads();

    // per-channel masked sums over this pixel chunk
    for (int c = 0; c < CCH; ++c) {
        const float* f = feats + ((size_t)b * CCH + c) * NPIX + base;
        float ac = 0.f, at = 0.f;
        for (int i = tid * 4; i < CHUNK; i += 1024) {
            v4f v = __builtin_nontemporal_load((const v4f*)(f + i));
            ac = fmaf(v.x, mk[i + 0], ac);
            ac = fmaf(v.y, mk[i + 1], ac);
            ac = fmaf(v.z, mk[i + 2], ac);
            ac = fmaf(v.w, mk[i + 3], ac);
            at += v.x + v.y + v.z + v.w;
        }
        float wc = wave_sum(ac), wt = wave_sum(at);
        if (lane == 0) { sC[wave] = wc; sT[wave] = wt; }
        __syncthreads();
        if (tid == 0) {
            float scv = 0.f, stv = 0.f;
            for (int i = 0; i < 8; ++i) { scv += sC[i]; stv += sT[i]; }
            atomicAdd(&pcs[b * CCH + c], scv);
            atomicAdd(&pns[b * CCH + c], stv - scv);  // non-change sum = total - change
        }
        __syncthreads();
    }
}

// ---------------- kernel 2: normalize prototypes, valid flag ----------------
// grid = B, block = 128 (one thread per channel)
__global__ void k_final(const float* __restrict__ pcs, const float* __restrict__ pns,
                        const float* __restrict__ ncs,
                        float* __restrict__ pcn, float* __restrict__ pnn,
                        float* __restrict__ validArr) {
    int b = blockIdx.x, c = threadIdx.x;
    int lane = c & 31, wave = c >> 5;
    __shared__ float rc[4], rn[4];

    float ncv = ncs[b];
    float nnv = (float)NPIX - ncv;
    float pc = pcs[b * CCH + c] / fmaxf(ncv, 1.f);
    float pn = pns[b * CCH + c] / fmaxf(nnv, 1.f);

    float wc = wave_sum(pc * pc), wn = wave_sum(pn * pn);
    if (lane == 0) { rc[wave] = wc; rn[wave] = wn; }
    __syncthreads();
    float n2c = rc[0] + rc[1] + rc[2] + rc[3];
    float n2n = rn[0] + rn[1] + rn[2] + rn[3];

    pcn[b * CCH + c] = pc / fmaxf(sqrtf(n2c), EPSF);
    pnn[b * CCH + c] = pn / fmaxf(sqrtf(n2n), EPSF);
    if (c == 0) validArr[b] = (ncv > 0.f && nnv > 0.f) ? 1.f : 0.f;
}

// ---------------- kernel 3: per-pixel sims via V_WMMA_F32_16X16X4_F32 -------
// One wave handles 16 pixels/tile: A = 16 pixels x 4 channels (fp32),
// B = 4 channels x 16 cols (col0 = p_c_norm, col1 = p_n_norm, rest 0).
// B fragments come from a 3-row LDS table (pc | pn | zeros) via a per-lane
// base pointer -> unconditional aligned ds_load_b64, no exec toggling.
// grid = (NPIX/1024, B), block = 256 (8 waves, 8 tiles each)
__global__ void k_sim(const float* __restrict__ feats, const int* __restrict__ gt,
                      const float* __restrict__ pcn, const float* __restrict__ pnn,
                      const float* __restrict__ validArr, float* __restrict__ out) {
    int b    = blockIdx.y;
    int tid  = threadIdx.x;
    int lane = tid & 31, wave = tid >> 5;
    int m = lane & 15, hi = lane >> 4;   // hi: 0 => K offset 0,1 ; 1 => K offset 2,3

    __shared__ __align__(16) float sB[3 * CCH];   // [pc | pn | zeros]
    __shared__ float sDC[8][16], sDN[8][16];

    if (tid < CCH) {
        sB[tid]           = pcn[b * CCH + tid];
        sB[2 * CCH + tid] = 0.f;                  // zero row for cols 2..15
    } else {
        sB[tid]           = pnn[b * CCH + (tid - CCH)];
    }
    __syncthreads();

    // per-lane B-row base: col0 -> pc row, col1 -> pn row, others -> zero row
    const float* bp = sB + ((m == 0) ? 0 : (m == 1) ? CCH : 2 * CCH);

    float vb = validArr[b];
    const float* fb = feats + (size_t)b * CCH * NPIX;
    const int*   gb = gt + (size_t)b * NPIX;
    float*       ob = out + (size_t)b * NPIX;

    int blockBase = blockIdx.x * 1024;
    for (int t = 0; t < 8; ++t) {
        int pix0 = blockBase + (wave * 8 + t) * 16;
        int pix  = pix0 + m;                 // A-matrix row (pixel) for this lane
        const float* fp = fb + pix;

        v8f  acc = {};
        float sq = 0.f;
        int hk = hi * 2;
#pragma unroll 4
        for (int k0 = 0; k0 < CCH; k0 += 4) {
            int c0 = k0 + hk;                // even -> 8B-aligned LDS access
            // A fragment: lane = m + 16*(k>=2), vgpr = k&1  (ISA 7.12.2, 32-bit A 16x4)
            float a0 = __builtin_nontemporal_load(fp + (size_t)c0 * NPIX);
            float a1 = __builtin_nontemporal_load(fp + (size_t)(c0 + 1) * NPIX);
            sq = fmaf(a0, a0, sq);
            sq = fmaf(a1, a1, sq);
            // B fragment: single aligned 64-bit LDS read, divergence-free
            v2f Bv = *reinterpret_cast<const v2f*>(bp + c0);
            v2f A  = {a0, a1};
            acc = __builtin_amdgcn_wmma_f32_16x16x4_f32(
                false, A, false, Bv, (short)0, acc, false, false);
        }

        // full ||f||^2 for pixel m: halves hold k%4 in {0,1} and {2,3}
        sq += __shfl_xor(sq, 16, 32);

        // D layout: acc[j] = D[j + 8*hi][lane&15]  -> bounce cols 0,1 through LDS
        if (m == 0) {
#pragma unroll
            for (int j = 0; j < 8; ++j) sDC[wave][hi * 8 + j] = acc[j];
        } else if (m == 1) {
#pragma unroll
            for (int j = 0; j < 8; ++j) sDN[wave][hi * 8 + j] = acc[j];
        }
        __syncthreads();

        if (lane < 16) {
            float dc  = sDC[wave][m];
            float dn  = sDN[wave][m];
            float inv = 1.f / fmaxf(sqrtf(sq), EPSF);
            float diff = (dc - dn) * inv;     // sim_pc - sim_pn
            int gv = gb[pix];
            float e = expf((gv == 1) ? diff : -diff);
            __builtin_nontemporal_store(vb * e, ob + pix);
        }
        __syncthreads();
    }
}

// ---------------- host launcher ---------------------------------------------
extern "C" void kernel_launch(void* const* d_in, const int* in_sizes, int n_in,
                              void* d_out, int out_size, void* d_ws, size_t ws_size,
                              hipStream_t stream) {
    const float* feats = (const float*)d_in[0];
    const int*   gt    = (const int*)d_in[1];
    float*       out   = (float*)d_out;
    float*       ws    = (float*)d_ws;

    // workspace layout (floats)
    float* pcs      = ws;          // [B*C]  change-class sums
    float* pns      = ws + 1024;   // [B*C]  non-change sums
    float* ncs      = ws + 2048;   // [B]    change-pixel counts
    float* pcn      = ws + 2056;   // [B*C]  normalized change prototype
    float* pnn      = ws + 3080;   // [B*C]  normalized non-change prototype
    float* validArr = ws + 4104;   // [B]

    k_zero<<<dim3(9), dim3(256), 0, stream>>>(ws, 2056);
    k_reduce<<<dim3(NPIX / 2048, NBATCH), dim3(256), 0, stream>>>(feats, gt, pcs, pns, ncs);
    k_final<<<dim3(NBATCH), dim3(128), 0, stream>>>(pcs, pns, ncs, pcn, pnn, validArr);
    k_sim<<<dim3(NPIX / 1024, NBATCH), dim3(256), 0, stream>>>(feats, gt, pcn, pnn, validArr, out);
}